// Derivative1D_31731218383126
// MI455X (gfx1250) — compile-verified
//
#include <hip/hip_runtime.h>

// Problem geometry (fixed by the reference): x = [64, 16384, 32] f32,
// y[b,i,c] = x[b,i+1,c] - x[b,i,c]  ->  flat per batch: out[j] = in[j+32] - in[j]
#define BATCHES      64
#define SEQ_L        16384
#define CHANS        32
#define IN_PER_B     (SEQ_L * CHANS)          // 524288 floats / batch
#define OUT_PER_B    ((SEQ_L - 1) * CHANS)    // 524256 floats / batch
#define HALO         CHANS                    // 32 floats
#define THREADS      256
#define TILE_OUT     4096                     // output floats per block-tile
#define TILES_PER_B  ((OUT_PER_B + TILE_OUT - 1) / TILE_OUT)  // 128

// Native clang vector type: valid for __builtin_nontemporal_store and still
// lowers to single b128 memory ops.
typedef float v4f __attribute__((ext_vector_type(4)));

__global__ __launch_bounds__(THREADS)
void Derivative1D_31731218383126_kernel(const float* __restrict__ in,
                                        float* __restrict__ out) {
    // Staging buffer: tile + halo (16.5 KB << 320 KB/WGP)
    __shared__ float lds[TILE_OUT + HALO];

    const int tile = blockIdx.x % TILES_PER_B;
    const int b    = blockIdx.x / TILES_PER_B;
    const int j0   = tile * TILE_OUT;                       // output offset in batch
    const int out_cnt = min(TILE_OUT, OUT_PER_B - j0);      // 4096 or 4064 (tail)
    // Staged input floats. Always ends exactly at or before the batch's input
    // end: full tile -> j0 + 4128 <= IN_PER_B; tail -> j0 + 4096 == IN_PER_B.
    const int in_cnt  = out_cnt + HALO;

    const float* __restrict__ src = in  + (size_t)b * IN_PER_B  + j0;
    float*       __restrict__ dst = out + (size_t)b * OUT_PER_B + j0;
    const int tid = threadIdx.x;

    // ---- Async stage: global -> LDS, 128-bit per lane, ASYNCcnt-tracked ----
    // LDS byte offset: low 32 bits of the generic pointer to a __shared__
    // object are the wave-relative LDS offset (flat LDS aperture mapping).
    const unsigned lds_base = (unsigned)(size_t)(void*)&lds[0];
    for (int c4 = tid; c4 * 4 < in_cnt; c4 += THREADS) {
        unsigned lds_addr = lds_base + (unsigned)(c4 * 16);
        const float* g = src + c4 * 4;
        asm volatile("global_load_async_to_lds_b128 %0, %1, off"
                     :
                     : "v"(lds_addr), "v"(g)
                     : "memory");
    }
    // Wait for this wave's async copies, then make all waves' LDS writes
    // visible across the workgroup.
    asm volatile("s_wait_asynccnt 0" ::: "memory");
    __syncthreads();

    // ---- Compute: b128 LDS reads (both 16B-aligned), NT b128 store ----
    for (int o4 = tid; o4 * 4 < out_cnt; o4 += THREADS) {
        const int j = o4 * 4;
        const v4f a = *(const v4f*)&lds[j];          // x[:, i, :]
        const v4f c = *(const v4f*)&lds[j + HALO];   // x[:, i+1, :]
        const v4f r = c - a;
        // Output is write-once, never re-read: bypass cache residency (NT),
        // preserving L2 capacity for the input halo reuse.
        __builtin_nontemporal_store(r, (v4f*)(dst + j));
    }
}

extern "C" void kernel_launch(void* const* d_in, const int* in_sizes, int n_in,
                              void* d_out, int out_size, void* d_ws, size_t ws_size,
                              hipStream_t stream) {
    (void)in_sizes; (void)n_in; (void)out_size; (void)d_ws; (void)ws_size;
    const float* x = (const float*)d_in[0];
    float*       y = (float*)d_out;

    dim3 grid(BATCHES * TILES_PER_B);   // 64 * 128 = 8192 blocks
    dim3 block(THREADS);                // 256 threads = 8 wave32
    Derivative1D_31731218383126_kernel<<<grid, block, 0, stream>>>(x, y);
}